// _PointnetSAModuleBase_1082331758729
// MI455X (gfx1250) — compile-verified
//
#include <hip/hip_runtime.h>

#define BATCH   8
#define NPTS    8192
#define NPOINT  2048
#define NSAMPLE 32
#define RADIUS2 0.01f   // 0.1^2

typedef __attribute__((ext_vector_type(2))) float v2f;
typedef __attribute__((ext_vector_type(8))) float v8f;

// ---------------------------------------------------------------------------
// Stage 1: Furthest Point Sampling. One workgroup per batch; each of 1024
// threads owns 8 points entirely in VGPRs. Per iteration: update min-dist,
// wave32 shuffle argmax, LDS cross-wave argmax, broadcast winner coords.
// Writes new_xyz (planar x/y/z) to workspace.
// ---------------------------------------------------------------------------
__global__ __launch_bounds__(1024) void fps_kernel(const float* __restrict__ xyz,
                                                   float* __restrict__ wsx,
                                                   float* __restrict__ wsy,
                                                   float* __restrict__ wsz) {
  __shared__ float red_d[32];
  __shared__ int   red_i[32];
  __shared__ float bcx, bcy, bcz;
  __shared__ int   bidx;

  const int b = blockIdx.x;
  const int t = threadIdx.x;
  const float* base = xyz + (size_t)b * NPTS * 3;

  float px[8], py[8], pz[8], dist[8];
#pragma unroll
  for (int k = 0; k < 8; ++k) {
    int j = t + (k << 10);
    px[k] = base[j * 3 + 0];
    py[k] = base[j * 3 + 1];
    pz[k] = base[j * 3 + 2];
    dist[k] = 1e10f;
  }
  if (t == 0) { bcx = px[0]; bcy = py[0]; bcz = pz[0]; bidx = 0; }
  __syncthreads();

  for (int i = 0; i < NPOINT; ++i) {
    float cx = bcx, cy = bcy, cz = bcz;
    if (t == 0) {  // emit current pick as new_xyz[b][i]
      wsx[b * NPOINT + i] = cx;
      wsy[b * NPOINT + i] = cy;
      wsz[b * NPOINT + i] = cz;
    }
    // update min-dist, local argmax (ascending k keeps smallest index on ties)
    float bd = -1.0f;
    int   bj = 0;
#pragma unroll
    for (int k = 0; k < 8; ++k) {
      float dx = px[k] - cx, dy = py[k] - cy, dz = pz[k] - cz;
      float d  = __builtin_fmaf(dx, dx, __builtin_fmaf(dy, dy, dz * dz));
      float nd = fminf(dist[k], d);
      dist[k]  = nd;
      if (nd > bd) { bd = nd; bj = t + (k << 10); }
    }
    // wave32 argmax (tie -> smaller index, matching jnp.argmax first-max)
#pragma unroll
    for (int off = 16; off > 0; off >>= 1) {
      float od = __shfl_xor(bd, off, 32);
      int   oj = __shfl_xor(bj, off, 32);
      if (od > bd || (od == bd && oj < bj)) { bd = od; bj = oj; }
    }
    if ((t & 31) == 0) { red_d[t >> 5] = bd; red_i[t >> 5] = bj; }
    __syncthreads();
    if (t < 32) {
      bd = red_d[t];
      bj = red_i[t];
#pragma unroll
      for (int off = 16; off > 0; off >>= 1) {
        float od = __shfl_xor(bd, off, 32);
        int   oj = __shfl_xor(bj, off, 32);
        if (od > bd || (od == bd && oj < bj)) { bd = od; bj = oj; }
      }
      if (t == 0) bidx = bj;
    }
    __syncthreads();
    const int widx = bidx;
    if (t == (widx & 1023)) {  // owner thread broadcasts winner coords
      const int ok = widx >> 10;
      float sx = px[0], sy = py[0], sz = pz[0];
#pragma unroll
      for (int k = 1; k < 8; ++k)
        if (ok == k) { sx = px[k]; sy = py[k]; sz = pz[k]; }
      bcx = sx; bcy = sy; bcz = sz;
    }
    __syncthreads();
  }
}

// ---------------------------------------------------------------------------
// Stage 2: Ball query + group + center + transpose.
// One wave per 16-center tile (8 batches * 128 tiles = 1024 waves).
// Per 16-point chunk: V_WMMA_F32_16X16X4_F32 computes the 16x16 dot tile
// (A = centers 16x4 zero-padded, B = points 4x16), epilogue forms the in-ball
// predicate (2*dot - |x|^2 >= |c|^2 - r^2, thresholds hoisted out of the
// loop so the hot loop has NO LDS traffic) and ballots a bitmask per D-VGPR.
// Lanes 0..15 each own one center row and append hits in ascending index
// order, then pad with the first hit.
// ---------------------------------------------------------------------------
__global__ __launch_bounds__(256) void ballquery_group_kernel(
    const float* __restrict__ xyz, const float* __restrict__ wsx,
    const float* __restrict__ wsy, const float* __restrict__ wsz,
    float* __restrict__ out) {
  const int gw   = (int)((blockIdx.x * 256u + threadIdx.x) >> 5);  // wave id
  const int lane = threadIdx.x & 31;
  const int b    = gw >> 7;       // 128 tiles per batch
  const int tile = gw & 127;
  const int r    = lane & 15;     // row (center) owned within tile
  const int m    = tile * 16 + r; // center index in [0, 2048)

  const float cx = wsx[b * NPOINT + m];
  const float cy = wsy[b * NPOINT + m];
  const float cz = wsz[b * NPOINT + m];
  // per-row threshold: in-ball  <=>  2*dot - |x|^2 >= |c|^2 - r^2
  const float cthr = cx * cx + cy * cy + cz * cz - RADIUS2;

  // A matrix 16x4 f32: lanes 0-15 hold K=0,1 ; lanes 16-31 hold K=2,3
  v2f A;
  A.x = (lane < 16) ? cx : cz;
  A.y = (lane < 16) ? cy : 0.0f;

  int cnt      = (lane < 16) ? 0 : NSAMPLE;  // inactive lanes report "done"
  int firstIdx = -1;
  const int hi = (lane < 16) ? 0 : 8;        // row-half offset for threshold bcast

  // Hoist the 8 row-threshold broadcasts (ds_bpermute) OUT of the chunk loop:
  // D VGPR v holds rows v (lanes 0-15) and v+8 (lanes 16-31); thresholds are
  // loop-invariant, so fetch them once.
  const float th0 = __shfl(cthr, 0 + hi, 32);
  const float th1 = __shfl(cthr, 1 + hi, 32);
  const float th2 = __shfl(cthr, 2 + hi, 32);
  const float th3 = __shfl(cthr, 3 + hi, 32);
  const float th4 = __shfl(cthr, 4 + hi, 32);
  const float th5 = __shfl(cthr, 5 + hi, 32);
  const float th6 = __shfl(cthr, 6 + hi, 32);
  const float th7 = __shfl(cthr, 7 + hi, 32);

  const float* pb = xyz + (size_t)b * NPTS * 3;
  float*       ob = out + (size_t)b * 3 * NPOINT * NSAMPLE;

  for (int c = 0; c < NPTS / 16; ++c) {
    const int n = c * 16 + r;  // lanes l and l+16 load the same column point
    float x = pb[n * 3 + 0];
    float y = pb[n * 3 + 1];
    float z = pb[n * 3 + 2];
    if (c + 4 < NPTS / 16)  // lead the loads by 4 chunks (192 B / lane-group)
      __builtin_prefetch(pb + (c + 4) * 48 + r * 3, 0, 1);
    const float xn = x * x + y * y + z * z;  // per-column |x|^2

    // B matrix 4x16 f32: lanes 0-15 hold K=0,1 ; lanes 16-31 hold K=2,3
    v2f Bm;
    Bm.x = (lane < 16) ? x : z;
    Bm.y = (lane < 16) ? y : 0.0f;

    v8f acc = {};
    acc = __builtin_amdgcn_wmma_f32_16x16x4_f32(false, A, false, Bm,
                                                (short)0, acc, false, false);

    // in-ball bitmask per D-VGPR: 2*dot - xn >= threshold(row)
    unsigned mk0, mk1, mk2, mk3, mk4, mk5, mk6, mk7;
#define BALL_MASK(vv, th, dst)                                               \
  {                                                                          \
    float lhs = __builtin_fmaf(2.0f, acc[vv], -xn);                          \
    dst = (unsigned)__ballot(lhs >= (th));                                   \
  }
    BALL_MASK(0, th0, mk0) BALL_MASK(1, th1, mk1)
    BALL_MASK(2, th2, mk2) BALL_MASK(3, th3, mk3)
    BALL_MASK(4, th4, mk4) BALL_MASK(5, th5, mk5)
    BALL_MASK(6, th6, mk6) BALL_MASK(7, th7, mk7)
#undef BALL_MASK

    // lane r extracts its row's 16-bit in-ball mask for this chunk
    const int v = r & 7;
    unsigned sel = (v < 4)
        ? ((v < 2) ? ((v == 0) ? mk0 : mk1) : ((v == 2) ? mk2 : mk3))
        : ((v < 6) ? ((v == 4) ? mk4 : mk5) : ((v == 6) ? mk6 : mk7));
    unsigned rm = (r < 8) ? (sel & 0xFFFFu) : (sel >> 16);

    if (lane < 16) {
      while (rm != 0u && cnt < NSAMPLE) {  // rarely taken: ~0.07 hits/chunk
        int bit = __builtin_ctz(rm);
        rm &= rm - 1u;
        int j = c * 16 + bit;
        if (firstIdx < 0) firstIdx = j;
        float qx = pb[j * 3 + 0] - cx;
        float qy = pb[j * 3 + 1] - cy;
        float qz = pb[j * 3 + 2] - cz;
        ob[(0 * NPOINT + m) * NSAMPLE + cnt] = qx;
        ob[(1 * NPOINT + m) * NSAMPLE + cnt] = qy;
        ob[(2 * NPOINT + m) * NSAMPLE + cnt] = qz;
        ++cnt;
      }
    }
    if ((unsigned)__ballot(cnt >= NSAMPLE) == 0xFFFFFFFFu) break;
  }

  // pad remaining slots with the first in-ball point (index 0 if none)
  if (lane < 16) {
    const int pad = (firstIdx < 0) ? 0 : firstIdx;
    float qx = pb[pad * 3 + 0] - cx;
    float qy = pb[pad * 3 + 1] - cy;
    float qz = pb[pad * 3 + 2] - cz;
    for (int s = cnt; s < NSAMPLE; ++s) {
      ob[(0 * NPOINT + m) * NSAMPLE + s] = qx;
      ob[(1 * NPOINT + m) * NSAMPLE + s] = qy;
      ob[(2 * NPOINT + m) * NSAMPLE + s] = qz;
    }
  }
}

// ---------------------------------------------------------------------------
extern "C" void kernel_launch(void* const* d_in, const int* in_sizes, int n_in,
                              void* d_out, int out_size, void* d_ws, size_t ws_size,
                              hipStream_t stream) {
  (void)in_sizes; (void)n_in; (void)out_size; (void)ws_size;
  const float* xyz = (const float*)d_in[0];
  float* wsx = (float*)d_ws;                 // new_xyz, planar x/y/z
  float* wsy = wsx + BATCH * NPOINT;
  float* wsz = wsy + BATCH * NPOINT;
  float* out = (float*)d_out;                // (B, 3, NPOINT, NSAMPLE) f32

  fps_kernel<<<BATCH, 1024, 0, stream>>>(xyz, wsx, wsy, wsz);
  ballquery_group_kernel<<<(BATCH * (NPOINT / 16) * 32) / 256, 256, 0, stream>>>(
      xyz, wsx, wsy, wsz, out);
}